// TransD_9251359555853
// MI455X (gfx1250) — compile-verified
//
#include <hip/hip_runtime.h>

// TransD margin-ranking loss, MI455X (gfx1250, wave32).
//
// Roofline: ~0.8 GB of gathered reads vs ~0.35 GFLOP -> HBM-bound (~34us
// floor at 23.3 TB/s). One wave32 = one triple pair (pos+neg); each 128-f32
// embedding row is one fully-coalesced 512B gather (32 lanes x b128), with
// all 12 row gathers issued back-to-back for deep MLP. All tables < 256MB,
// so gathers use SGPR-base + 32-bit VGPR byte offsets (no 64-bit VALU math).
//
// Wave allreduces (2 dot-diffs + 2 squared norms) run on the matrix pipe:
// V_WMMA_F32_16X16X4_F32 with an all-ones B computes row sums of A
// (D[i][j] = sum_k A[i,k]); summing the 8 D VGPRs per lane + one
// shfl_xor(16) yields the sum over all 64 A slots -- invariant to A/B
// storage-layout permutations, f32-exact, fixed order (deterministic).
//
// Final loss: per-block partials in d_ws, then a single fixed-order
// reduction block (no float atomics -> bitwise deterministic).

#define EPS_F    1e-6f
#define MARGIN_F 1.0f

typedef __attribute__((ext_vector_type(2))) float v2f;
typedef __attribute__((ext_vector_type(8))) float v8f;

__device__ __forceinline__ float4 ld4_off(const float* __restrict__ base,
                                          unsigned byte_off) {
    return *(const float4*)((const char*)base + byte_off);
}

__device__ __forceinline__ float dot4(const float4 a, const float4 b) {
    return a.x * b.x + a.y * b.y + a.z * b.z + a.w * b.w;
}

// Wave32 allreduce via the matrix pipe (see header comment).
__device__ __forceinline__ float wave_allreduce_wmma(float v) {
    v2f a;    a[0] = v;    a[1] = 0.0f;     // A: 16x4, our 32 values + zeros
    v2f ones; ones[0] = 1.0f; ones[1] = 1.0f; // B: all-ones 4x16 (layout-proof)
    v8f c = {};                               // C = 0
    v8f d = __builtin_amdgcn_wmma_f32_16x16x4_f32(
        /*neg_a=*/false, a, /*neg_b=*/false, ones,
        /*c_mod=*/(short)0, c, /*reuse_a=*/false, /*reuse_b=*/false);
    float h = ((d[0] + d[1]) + (d[2] + d[3])) + ((d[4] + d[5]) + (d[6] + d[7]));
    return h + __shfl_xor(h, 16, 32);         // combine the two row halves
}

__global__ void __launch_bounds__(256) transd_partial_kernel(
    const int*   __restrict__ posX,   // (B,3) int32: h,r,t
    const int*   __restrict__ negX,   // (B,3)
    const float* __restrict__ W_e,    // (NE,128)
    const float* __restrict__ W_ep,   // (NE,128)
    const float* __restrict__ W_r,    // (NR,128)
    const float* __restrict__ W_rp,   // (NR,128)
    float*       __restrict__ ws,     // per-block partials
    int B)
{
    __shared__ float wave_loss[8];

    const int wave = threadIdx.x >> 5;   // 8 waves per block (wave32)
    const int lane = threadIdx.x & 31;
    const int b    = blockIdx.x * 8 + wave;

    float loss = 0.0f;
    if (b < B) {                          // wave-uniform guard (EXEC all-ones)
        // ---- indices (uniform within the wave; broadcast via cache) ----
        const int hp = posX[3 * b + 0];
        const int rp = posX[3 * b + 1];
        const int tp = posX[3 * b + 2];
        const int hn = negX[3 * b + 0];
        const int rn = negX[3 * b + 1];
        const int tn = negX[3 * b + 2];

        // 32-bit byte offsets: row*512B + lane*16B (all tables < 256MB)
        const unsigned lo  = (unsigned)lane << 4;
        const unsigned oHP = ((unsigned)hp << 9) + lo;
        const unsigned oTP = ((unsigned)tp << 9) + lo;
        const unsigned oRP = ((unsigned)rp << 9) + lo;
        const unsigned oHN = ((unsigned)hn << 9) + lo;
        const unsigned oTN = ((unsigned)tn << 9) + lo;
        const unsigned oRN = ((unsigned)rn << 9) + lo;

        // ---- issue all 12 b128 gathers back-to-back (deep MLP) ----
        const float4 PH  = ld4_off(W_e,  oHP);
        const float4 PHp = ld4_off(W_ep, oHP);
        const float4 PT  = ld4_off(W_e,  oTP);
        const float4 PTp = ld4_off(W_ep, oTP);
        const float4 PR  = ld4_off(W_r,  oRP);
        const float4 PRp = ld4_off(W_rp, oRP);

        const float4 NH  = ld4_off(W_e,  oHN);
        const float4 NHp = ld4_off(W_ep, oHN);
        const float4 NT  = ld4_off(W_e,  oTN);
        const float4 NTp = ld4_off(W_ep, oTN);
        const float4 NR_ = ld4_off(W_r,  oRN);
        const float4 NRp = ld4_off(W_rp, oRN);

        // ---- per-lane dot difference: dot(headp,head) - dot(tailp,tail) ----
        // (only the difference is needed: sum(a-b) == sum(a)-sum(b))
        const float pdiff = dot4(PHp, PH) - dot4(PTp, PT);
        const float ndiff = dot4(NHp, NH) - dot4(NTp, NT);

        const float pd = wave_allreduce_wmma(pdiff);
        const float nd = wave_allreduce_wmma(ndiff);

        // diff = (head - tail) + relp*(s_h - s_t) + rel + EPS
        float qp, qn;
        {
            float dx = (PH.x - PT.x) + PRp.x * pd + PR.x + EPS_F;
            float dy = (PH.y - PT.y) + PRp.y * pd + PR.y + EPS_F;
            float dz = (PH.z - PT.z) + PRp.z * pd + PR.z + EPS_F;
            float dw = (PH.w - PT.w) + PRp.w * pd + PR.w + EPS_F;
            qp = dx * dx + dy * dy + dz * dz + dw * dw;
        }
        {
            float dx = (NH.x - NT.x) + NRp.x * nd + NR_.x + EPS_F;
            float dy = (NH.y - NT.y) + NRp.y * nd + NR_.y + EPS_F;
            float dz = (NH.z - NT.z) + NRp.z * nd + NR_.z + EPS_F;
            float dw = (NH.w - NT.w) + NRp.w * nd + NR_.w + EPS_F;
            qn = dx * dx + dy * dy + dz * dz + dw * dw;
        }

        const float sqp = wave_allreduce_wmma(qp);
        const float sqn = wave_allreduce_wmma(qn);

        loss = fmaxf(sqrtf(sqp) - sqrtf(sqn) + MARGIN_F, 0.0f);
    }

    if (lane == 0) wave_loss[wave] = loss;   // loss == 0 for tail waves
    __syncthreads();

    // deterministic fixed-order sum of the 8 wave results
    if (threadIdx.x == 0) {
        float s = 0.0f;
        #pragma unroll
        for (int i = 0; i < 8; ++i) s += wave_loss[i];
        ws[blockIdx.x] = s;
    }
}

// Single-block deterministic reduction of per-block partials.
__global__ void __launch_bounds__(256) transd_reduce_kernel(
    const float* __restrict__ ws, float* __restrict__ out, int n, float inv_b)
{
    __shared__ float smem[256];
    float acc = 0.0f;
    for (int i = threadIdx.x; i < n; i += 256) acc += ws[i];  // fixed order
    smem[threadIdx.x] = acc;
    __syncthreads();
    #pragma unroll
    for (int m = 128; m > 0; m >>= 1) {
        if (threadIdx.x < m) smem[threadIdx.x] += smem[threadIdx.x + m];
        __syncthreads();
    }
    if (threadIdx.x == 0) out[0] = smem[0] * inv_b;
}

extern "C" void kernel_launch(void* const* d_in, const int* in_sizes, int n_in,
                              void* d_out, int out_size, void* d_ws, size_t ws_size,
                              hipStream_t stream) {
    const int*   posX = (const int*)d_in[0];
    const int*   negX = (const int*)d_in[1];
    const float* W_e  = (const float*)d_in[2];
    const float* W_ep = (const float*)d_in[3];
    const float* W_r  = (const float*)d_in[4];
    const float* W_rp = (const float*)d_in[5];
    float* out = (float*)d_out;
    float* ws  = (float*)d_ws;

    const int B = in_sizes[0] / 3;            // 131072
    const int blocks = (B + 7) / 8;           // 8 waves (triples) per block

    transd_partial_kernel<<<blocks, 256, 0, stream>>>(
        posX, negX, W_e, W_ep, W_r, W_rp, ws, B);
    transd_reduce_kernel<<<1, 256, 0, stream>>>(
        ws, out, blocks, 1.0f / (float)B);
    (void)n_in; (void)out_size; (void)ws_size;
}